// G2Net_21285857919253
// MI455X (gfx1250) — compile-verified
//
#include <hip/hip_runtime.h>
#include <math.h>

#define BNB 2
#define NC 19
#define CP 32
#define HH 64
#define WW 64
#define NN 4096
#define CFD 256

typedef float v2f __attribute__((ext_vector_type(2)));
typedef float v8f __attribute__((ext_vector_type(8)));

__device__ __forceinline__ v8f wmma4(v2f a, v2f b, v8f c) {
#if defined(__gfx1250__)
  // D = A(16x4 f32) * B(4x16 f32) + C(16x16 f32)
  return __builtin_amdgcn_wmma_f32_16x16x4_f32(false, a, false, b, (short)0, c,
                                               false, false);
#else
  (void)a; (void)b;
  return c;  // host-pass stub, never executed
#endif
}

// ---------------------------------------------------------------------------
// K0: softmax over classes, confidence mask, argmax, padded pred^T [B,N,32]
// ---------------------------------------------------------------------------
__global__ void k_prep(const float* __restrict__ pred, const float* __restrict__ cls_w,
                       float* __restrict__ predT, float* __restrict__ maskb,
                       int* __restrict__ pmaxb) {
  int idx = blockIdx.x * blockDim.x + threadIdx.x;
  if (idx >= BNB * NN) return;
  int b = idx / NN, n = idx % NN;
  float v[NC];
  float mx = -1e30f;
#pragma unroll
  for (int c = 0; c < NC; c++) {
    v[c] = pred[(size_t)(b * NC + c) * NN + n];
    mx = fmaxf(mx, v[c]);
  }
  float s = 0.f;
#pragma unroll
  for (int c = 0; c < NC; c++) s += expf(v[c] - mx);
  int am = 0;
  float bv = v[0];
#pragma unroll
  for (int c = 1; c < NC; c++)
    if (v[c] > bv) { bv = v[c]; am = c; }
  float pprob = expf(bv - mx) / s;
  float cwmax = 0.f;
#pragma unroll
  for (int c = 0; c < NC; c++) cwmax = fmaxf(cwmax, cls_w[c]);
  float cwm = cls_w[am] / (cwmax + 1e-10f) * 0.95f;
  maskb[idx] = (pprob >= 0.95f || pprob >= cwm) ? 1.f : 0.f;
  pmaxb[idx] = am;
  float* pt = predT + (size_t)idx * CP;
#pragma unroll
  for (int c = 0; c < NC; c++) pt[c] = v[c];
#pragma unroll
  for (int c = NC; c < CP; c++) pt[c] = 0.f;
}

// ---------------------------------------------------------------------------
// K1: transpose feat [B,256,N] -> featT [B,N,256] via 32x32 LDS tiles
// ---------------------------------------------------------------------------
__global__ void k_transpose_feat(const float* __restrict__ feat,
                                 float* __restrict__ featT) {
  __shared__ float tile[32][33];
  int b = blockIdx.z;
  int n0 = blockIdx.x * 32, c0 = blockIdx.y * 32;
  for (int i = threadIdx.y; i < 32; i += 8)
    tile[i][threadIdx.x] = feat[(size_t)(b * CFD + c0 + i) * NN + n0 + threadIdx.x];
  __syncthreads();
  for (int i = threadIdx.y; i < 32; i += 8)
    featT[(size_t)(b * NN + n0 + i) * CFD + c0 + threadIdx.x] = tile[threadIdx.x][i];
}

// ---------------------------------------------------------------------------
// K2: row-normalize featT -> xnT (x / (||x|| + 1e-9))
// ---------------------------------------------------------------------------
__global__ void k_rownorm(const float* __restrict__ featT, float* __restrict__ xnT) {
  __shared__ float red[64];
  int row = blockIdx.x;
  const float* src = featT + (size_t)row * CFD;
  float s = 0.f;
  for (int c = threadIdx.x; c < CFD; c += 64) { float x = src[c]; s += x * x; }
  red[threadIdx.x] = s;
  __syncthreads();
  for (int off = 32; off > 0; off >>= 1) {
    if (threadIdx.x < off) red[threadIdx.x] += red[threadIdx.x + off];
    __syncthreads();
  }
  float inv = 1.f / (sqrtf(red[0]) + 1e-9f);
  float* dst = xnT + (size_t)row * CFD;
  for (int c = threadIdx.x; c < CFD; c += 64) dst[c] = src[c] * inv;
}

// ---------------------------------------------------------------------------
// K3: Gram matrix with clamp: OUT = max(X X^T, 0), X = [N,256] per batch.
// Block = 128 thr (4 waves), 64x64 tile, each wave 2x2 of 16x16 WMMA tiles.
// ---------------------------------------------------------------------------
__global__ void k_gram_clamp(const float* __restrict__ X, float* __restrict__ OUT) {
  __shared__ float As[64 * 34];
  __shared__ float Bs[64 * 34];
  const int b = blockIdx.z;
  const int i0 = blockIdx.x * 64, j0 = blockIdx.y * 64;
  const float* Xb = X + (size_t)b * NN * CFD;
  const int t = threadIdx.x;
  const int lane = t & 31, wave = t >> 5;
  const int half = lane >> 4, l16 = lane & 15;
  const int wm = (wave >> 1) * 32, wn = (wave & 1) * 32;
  v8f z = {0.f, 0.f, 0.f, 0.f, 0.f, 0.f, 0.f, 0.f};
  v8f acc[2][2];
  acc[0][0] = z; acc[0][1] = z; acc[1][0] = z; acc[1][1] = z;
  for (int k0 = 0; k0 < CFD; k0 += 32) {
    __syncthreads();
    for (int lin = t; lin < 2048; lin += 128) {
      int r = lin >> 5, k = lin & 31;
      As[r * 34 + k] = Xb[(size_t)(i0 + r) * CFD + k0 + k];
      Bs[r * 34 + k] = Xb[(size_t)(j0 + r) * CFD + k0 + k];
    }
    __syncthreads();
#pragma unroll
    for (int kc = 0; kc < 32; kc += 4) {
#pragma unroll
      for (int ti = 0; ti < 2; ti++) {
        v2f a;
        a.x = As[(wm + ti * 16 + l16) * 34 + kc + half * 2];
        a.y = As[(wm + ti * 16 + l16) * 34 + kc + half * 2 + 1];
#pragma unroll
        for (int tj = 0; tj < 2; tj++) {
          v2f bb;
          bb.x = Bs[(wn + tj * 16 + l16) * 34 + kc + half * 2];
          bb.y = Bs[(wn + tj * 16 + l16) * 34 + kc + half * 2 + 1];
          acc[ti][tj] = wmma4(a, bb, acc[ti][tj]);
        }
      }
    }
  }
  float* Ob = OUT + (size_t)b * NN * NN;
#pragma unroll
  for (int ti = 0; ti < 2; ti++)
#pragma unroll
    for (int tj = 0; tj < 2; tj++)
#pragma unroll
      for (int r = 0; r < 8; r++) {
        int gi = i0 + wm + ti * 16 + r + half * 8;
        int gj = j0 + wn + tj * 16 + l16;
        Ob[(size_t)gi * NN + gj] = fmaxf(acc[ti][tj][r], 0.f);
      }
}

// ---------------------------------------------------------------------------
// K4: exact per-row 21st-largest via radix-select on float bits (vals >= 0),
// threshold row in place (v < thr -> 0), emit row sum.
// ---------------------------------------------------------------------------
__global__ void k_topk_threshold(float* __restrict__ T, float* __restrict__ rowsum) {
  __shared__ float row[NN];
  __shared__ unsigned int hist[256];
  __shared__ float red[256];
  __shared__ unsigned int s_prefix;
  __shared__ int s_k;
  int r = blockIdx.x;  // b*N + i
  float* rp = T + (size_t)r * NN;
  int t = threadIdx.x;
  for (int j = t; j < NN; j += 256) row[j] = rp[j];
  if (t == 0) { s_prefix = 0u; s_k = 21; }  // 21st largest == sort-desc[20]
  __syncthreads();
  for (int byte = 3; byte >= 0; byte--) {
    hist[t] = 0u;
    __syncthreads();
    unsigned int pre = s_prefix;
    unsigned int premask = (byte == 3) ? 0u : (0xFFFFFFFFu << ((byte + 1) * 8));
    for (int j = t; j < NN; j += 256) {
      unsigned int u = __float_as_uint(row[j]);
      if ((u & premask) == pre) atomicAdd(&hist[(u >> (byte * 8)) & 255], 1u);
    }
    __syncthreads();
    if (t == 0) {
      int k = s_k;
      int bin = 255;
      for (; bin >= 0; bin--) {
        int c = (int)hist[bin];
        if (k - c <= 0) break;
        k -= c;
      }
      if (bin < 0) bin = 0;
      s_prefix = pre | ((unsigned int)bin << (byte * 8));
      s_k = k;
    }
    __syncthreads();
  }
  float thr = __uint_as_float(s_prefix);
  float s = 0.f;
  for (int j = t; j < NN; j += 256) {
    float v = row[j];
    v = (v < thr) ? 0.f : v;
    rp[j] = v;
    s += v;
  }
  red[t] = s;
  __syncthreads();
  for (int off = 128; off > 0; off >>= 1) {
    if (t < off) red[t] += red[t + off];
    __syncthreads();
  }
  if (t == 0) rowsum[r] = red[0];
}

// ---------------------------------------------------------------------------
// K5: column sums of T; d1 = rsqrt(rowsum + colsum + 1e-10)  (A = T + T^T)
// ---------------------------------------------------------------------------
__global__ void k_colsum_d1(const float* __restrict__ T, const float* __restrict__ rowsum,
                            float* __restrict__ d1) {
  int idx = blockIdx.x * blockDim.x + threadIdx.x;
  if (idx >= BNB * NN) return;
  int b = idx / NN, j = idx % NN;
  const float* Tb = T + (size_t)b * NN * NN;
  float s = 0.f;
  for (int i = 0; i < NN; i++) s += Tb[(size_t)i * NN + j];
  d1[idx] = rsqrtf(rowsum[idx] + s + 1e-10f);
}

// ---------------------------------------------------------------------------
// K6: p2 = d1 .* ((T + T^T) @ (d1 .* predT)), blended with mask -> p3T.
// One wave per 16-row block; symmetrized A tile built in LDS, then WMMA.
// ---------------------------------------------------------------------------
__global__ void k_prop_pred(const float* __restrict__ T, const float* __restrict__ predT,
                            const float* __restrict__ d1, const float* __restrict__ maskb,
                            float* __restrict__ p3T) {
  __shared__ float Ac[16 * 68];  // (T_ij + T_ji^T) tile, 16x64
  __shared__ float Bt[64 * 17];  // T_ji raw, 64x16
  __shared__ float Yt[64 * 34];  // d1-scaled predT chunk, 64x32
  const int b = blockIdx.y;
  const int i0 = blockIdx.x * 16;
  const float* Tb = T + (size_t)b * NN * NN;
  const int lane = threadIdx.x;
  const int half = lane >> 4, l16 = lane & 15;
  v8f z = {0.f, 0.f, 0.f, 0.f, 0.f, 0.f, 0.f, 0.f};
  v8f acc0 = z, acc1 = z;
  for (int j0 = 0; j0 < NN; j0 += 64) {
    __syncthreads();
    for (int lin = lane; lin < 1024; lin += 32) {
      int m = lin >> 6, k = lin & 63;
      Ac[m * 68 + k] = Tb[(size_t)(i0 + m) * NN + j0 + k];
    }
    for (int lin = lane; lin < 1024; lin += 32) {
      int k = lin >> 4, m = lin & 15;
      Bt[k * 17 + m] = Tb[(size_t)(j0 + k) * NN + i0 + m];
    }
    for (int lin = lane; lin < 2048; lin += 32) {
      int k = lin >> 5, c = lin & 31;
      Yt[k * 34 + c] = d1[b * NN + j0 + k] * predT[(size_t)(b * NN + j0 + k) * CP + c];
    }
    __syncthreads();
    for (int lin = lane; lin < 1024; lin += 32) {
      int m = lin >> 6, k = lin & 63;
      Ac[m * 68 + k] += Bt[k * 17 + m];
    }
    __syncthreads();
#pragma unroll
    for (int kc = 0; kc < 64; kc += 4) {
      v2f a;
      a.x = Ac[l16 * 68 + kc + half * 2];
      a.y = Ac[l16 * 68 + kc + half * 2 + 1];
      v2f b0;
      b0.x = Yt[(kc + half * 2) * 34 + l16];
      b0.y = Yt[(kc + half * 2 + 1) * 34 + l16];
      acc0 = wmma4(a, b0, acc0);
      v2f b1;
      b1.x = Yt[(kc + half * 2) * 34 + 16 + l16];
      b1.y = Yt[(kc + half * 2 + 1) * 34 + 16 + l16];
      acc1 = wmma4(a, b1, acc1);
    }
  }
#pragma unroll
  for (int r = 0; r < 8; r++) {
    int gi = i0 + r + half * 8;
    int gidx = b * NN + gi;
    float di = d1[gidx], mk = maskb[gidx];
    float wA = 0.8f * (1.f - mk) + 0.2f * mk;  // weight on propagated p2
    float wB = 0.2f * (1.f - mk) + 0.8f * mk;  // weight on original pred
    {
      int c = l16;
      float p2 = acc0[r] * di;
      float pr = predT[(size_t)gidx * CP + c];
      p3T[(size_t)gidx * CP + c] = wA * p2 + wB * pr;
    }
    {
      int c = 16 + l16;
      float p2 = acc1[r] * di;
      float pr = predT[(size_t)gidx * CP + c];
      p3T[(size_t)gidx * CP + c] = wA * p2 + wB * pr;
    }
  }
}

// ---------------------------------------------------------------------------
// K7: 19x19 double conv head -> pred_out; softmax + row-normalize -> spnT
// ---------------------------------------------------------------------------
__global__ void k_cls_head(const float* __restrict__ p3T, const float* __restrict__ w1,
                           const float* __restrict__ b1, const float* __restrict__ w2,
                           const float* __restrict__ b2, float* __restrict__ predOut,
                           float* __restrict__ spnT) {
  int idx = blockIdx.x * blockDim.x + threadIdx.x;
  if (idx >= BNB * NN) return;
  int b = idx / NN, n = idx % NN;
  float x[NC], tv[NC], y[NC];
#pragma unroll
  for (int c = 0; c < NC; c++) x[c] = p3T[(size_t)idx * CP + c];
  for (int o = 0; o < NC; o++) {
    float s = b1[o];
#pragma unroll
    for (int c = 0; c < NC; c++) s += w1[o * NC + c] * x[c];
    tv[o] = s;
  }
  for (int o = 0; o < NC; o++) {
    float s = b2[o];
#pragma unroll
    for (int c = 0; c < NC; c++) s += w2[o * NC + c] * tv[c];
    y[o] = s;
  }
#pragma unroll
  for (int o = 0; o < NC; o++) predOut[(size_t)(b * NC + o) * NN + n] = y[o];
  float mx = y[0];
#pragma unroll
  for (int o = 1; o < NC; o++) mx = fmaxf(mx, y[o]);
  float sp[NC];
  float s = 0.f;
#pragma unroll
  for (int o = 0; o < NC; o++) { sp[o] = expf(y[o] - mx); s += sp[o]; }
  float inv_s = 1.f / s;
  float nrm = 0.f;
#pragma unroll
  for (int o = 0; o < NC; o++) { sp[o] *= inv_s; nrm += sp[o] * sp[o]; }
  float inv = 1.f / (sqrtf(nrm) + 1e-9f);
  float* dst = spnT + (size_t)idx * CP;
#pragma unroll
  for (int o = 0; o < NC; o++) dst[o] = sp[o] * inv;
#pragma unroll
  for (int o = NC; o < CP; o++) dst[o] = 0.f;
}

// ---------------------------------------------------------------------------
// K8: scrambled one-hot rows (faithful to torch permute(0,2,3,1).view bug)
// row n, slot k -> flat f = n*19+k over [W,C,H] layout of onehot(p_max)
// ---------------------------------------------------------------------------
__global__ void k_oh(const int* __restrict__ pmaxb, float* __restrict__ ohT) {
  int idx = blockIdx.x * blockDim.x + threadIdx.x;
  if (idx >= BNB * NN * CP) return;
  int k = idx % CP;
  int n = (idx / CP) % NN;
  int b = idx / (CP * NN);
  float val = 0.f;
  if (k < NC) {
    int f = n * NC + k;
    int w = f / (NC * HH);
    int rem = f % (NC * HH);
    int c = rem / HH;
    int h = rem % HH;
    val = (pmaxb[b * NN + h * WW + w] == c) ? 1.f : 0.f;
  }
  ohT[idx] = val;
}

// ---------------------------------------------------------------------------
// K9: aff_f = (oh-dot > 0) ? max(cos_sp, 0) : 0 ; two Grams fused, K=32
// ---------------------------------------------------------------------------
__global__ void k_afff(const float* __restrict__ spnT, const float* __restrict__ ohT,
                       float* __restrict__ OUT) {
  __shared__ float As[64 * 34];
  __shared__ float Bs[64 * 34];
  __shared__ float Ao[64 * 34];
  __shared__ float Bo[64 * 34];
  const int b = blockIdx.z;
  const int i0 = blockIdx.x * 64, j0 = blockIdx.y * 64;
  const float* Sb = spnT + (size_t)b * NN * CP;
  const float* Hb = ohT + (size_t)b * NN * CP;
  const int t = threadIdx.x;
  const int lane = t & 31, wave = t >> 5;
  const int half = lane >> 4, l16 = lane & 15;
  const int wm = (wave >> 1) * 32, wn = (wave & 1) * 32;
  for (int lin = t; lin < 2048; lin += 128) {
    int r = lin >> 5, k = lin & 31;
    As[r * 34 + k] = Sb[(size_t)(i0 + r) * CP + k];
    Bs[r * 34 + k] = Sb[(size_t)(j0 + r) * CP + k];
    Ao[r * 34 + k] = Hb[(size_t)(i0 + r) * CP + k];
    Bo[r * 34 + k] = Hb[(size_t)(j0 + r) * CP + k];
  }
  __syncthreads();
  v8f z = {0.f, 0.f, 0.f, 0.f, 0.f, 0.f, 0.f, 0.f};
  v8f accS[2][2], accO[2][2];
  accS[0][0] = z; accS[0][1] = z; accS[1][0] = z; accS[1][1] = z;
  accO[0][0] = z; accO[0][1] = z; accO[1][0] = z; accO[1][1] = z;
#pragma unroll
  for (int kc = 0; kc < 32; kc += 4) {
#pragma unroll
    for (int ti = 0; ti < 2; ti++) {
      v2f aS, aO;
      aS.x = As[(wm + ti * 16 + l16) * 34 + kc + half * 2];
      aS.y = As[(wm + ti * 16 + l16) * 34 + kc + half * 2 + 1];
      aO.x = Ao[(wm + ti * 16 + l16) * 34 + kc + half * 2];
      aO.y = Ao[(wm + ti * 16 + l16) * 34 + kc + half * 2 + 1];
#pragma unroll
      for (int tj = 0; tj < 2; tj++) {
        v2f bS, bO;
        bS.x = Bs[(wn + tj * 16 + l16) * 34 + kc + half * 2];
        bS.y = Bs[(wn + tj * 16 + l16) * 34 + kc + half * 2 + 1];
        bO.x = Bo[(wn + tj * 16 + l16) * 34 + kc + half * 2];
        bO.y = Bo[(wn + tj * 16 + l16) * 34 + kc + half * 2 + 1];
        accS[ti][tj] = wmma4(aS, bS, accS[ti][tj]);
        accO[ti][tj] = wmma4(aO, bO, accO[ti][tj]);
      }
    }
  }
  float* Ob = OUT + (size_t)b * NN * NN;
#pragma unroll
  for (int ti = 0; ti < 2; ti++)
#pragma unroll
    for (int tj = 0; tj < 2; tj++)
#pragma unroll
      for (int r = 0; r < 8; r++) {
        int gi = i0 + wm + ti * 16 + r + half * 8;
        int gj = j0 + wn + tj * 16 + l16;
        float od = accO[ti][tj][r];
        Ob[(size_t)gi * NN + gj] = (od > 0.f) ? fmaxf(accS[ti][tj][r], 0.f) : 0.f;
      }
}

// ---------------------------------------------------------------------------
// K10: row sums of (symmetric) AF -> d2 = rsqrt(sum + 1e-10)
// ---------------------------------------------------------------------------
__global__ void k_rowsum_rsqrt(const float* __restrict__ M, float* __restrict__ dout) {
  __shared__ float red[256];
  int r = blockIdx.x;
  const float* rp = M + (size_t)r * NN;
  int t = threadIdx.x;
  float s = 0.f;
  for (int j = t; j < NN; j += 256) s += rp[j];
  red[t] = s;
  __syncthreads();
  for (int off = 128; off > 0; off >>= 1) {
    if (t < off) red[t] += red[t + off];
    __syncthreads();
  }
  if (t == 0) dout[r] = rsqrtf(red[0] + 1e-10f);
}

// ---------------------------------------------------------------------------
// K11: new_feat = d2 .* (AF @ (d2 .* featT))   [N,N] x [N,256]
// ---------------------------------------------------------------------------
__global__ void k_prop_feat(const float* __restrict__ AF, const float* __restrict__ featT,
                            const float* __restrict__ d2, float* __restrict__ nfT) {
  __shared__ float As[64 * 34];  // AF rows 64 x k32
  __shared__ float Bs[32 * 68];  // scaled feat chunk k32 x n64
  const int b = blockIdx.z;
  const int i0 = blockIdx.x * 64, c0 = blockIdx.y * 64;
  const float* Ab = AF + (size_t)b * NN * NN;
  const int t = threadIdx.x;
  const int lane = t & 31, wave = t >> 5;
  const int half = lane >> 4, l16 = lane & 15;
  const int wm = (wave >> 1) * 32, wn = (wave & 1) * 32;
  v8f z = {0.f, 0.f, 0.f, 0.f, 0.f, 0.f, 0.f, 0.f};
  v8f acc[2][2];
  acc[0][0] = z; acc[0][1] = z; acc[1][0] = z; acc[1][1] = z;
  for (int m0 = 0; m0 < NN; m0 += 32) {
    __syncthreads();
    for (int lin = t; lin < 2048; lin += 128) {
      int r = lin >> 5, k = lin & 31;
      As[r * 34 + k] = Ab[(size_t)(i0 + r) * NN + m0 + k];
      int kk = lin >> 6, c = lin & 63;
      Bs[kk * 68 + c] = d2[b * NN + m0 + kk] *
                        featT[(size_t)(b * NN + m0 + kk) * CFD + c0 + c];
    }
    __syncthreads();
#pragma unroll
    for (int kc = 0; kc < 32; kc += 4) {
#pragma unroll
      for (int ti = 0; ti < 2; ti++) {
        v2f a;
        a.x = As[(wm + ti * 16 + l16) * 34 + kc + half * 2];
        a.y = As[(wm + ti * 16 + l16) * 34 + kc + half * 2 + 1];
#pragma unroll
        for (int tj = 0; tj < 2; tj++) {
          v2f bb;
          bb.x = Bs[(kc + half * 2) * 68 + wn + tj * 16 + l16];
          bb.y = Bs[(kc + half * 2 + 1) * 68 + wn + tj * 16 + l16];
          acc[ti][tj] = wmma4(a, bb, acc[ti][tj]);
        }
      }
    }
  }
#pragma unroll
  for (int ti = 0; ti < 2; ti++)
#pragma unroll
    for (int tj = 0; tj < 2; tj++)
#pragma unroll
      for (int r = 0; r < 8; r++) {
        int gi = i0 + wm + ti * 16 + r + half * 8;
        int gj = c0 + wn + tj * 16 + l16;
        nfT[(size_t)(b * NN + gi) * CFD + gj] = acc[ti][tj][r] * d2[b * NN + gi];
      }
}

// ---------------------------------------------------------------------------
// K12/K13: dense projection OUT = act(X @ W^T + bias); optional transposed
// store to [B,256,H,W] via LDS re-tiling for coalesced writes.
// ---------------------------------------------------------------------------
template <bool RELU, bool TRANSOUT>
__global__ void k_dense(const float* __restrict__ X, const float* __restrict__ W,
                        const float* __restrict__ bias, float* __restrict__ OUT) {
  __shared__ float smem[64 * 34 + 32 * 68];  // 4352 floats, reused for C-tile
  float* As = smem;            // 64 x 34  (X rows)
  float* Bsm = smem + 64 * 34; // 32 x 68  (W^T chunk)
  const int b = blockIdx.z;
  const int n0 = blockIdx.x * 64, o0 = blockIdx.y * 64;
  const float* Xb = X + (size_t)b * NN * CFD;
  const int t = threadIdx.x;
  const int lane = t & 31, wave = t >> 5;
  const int half = lane >> 4, l16 = lane & 15;
  const int wm = (wave >> 1) * 32, wn = (wave & 1) * 32;
  v8f z = {0.f, 0.f, 0.f, 0.f, 0.f, 0.f, 0.f, 0.f};
  v8f acc[2][2];
  acc[0][0] = z; acc[0][1] = z; acc[1][0] = z; acc[1][1] = z;
  for (int k0 = 0; k0 < CFD; k0 += 32) {
    __syncthreads();
    for (int lin = t; lin < 2048; lin += 128) {
      int r = lin >> 5, k = lin & 31;
      As[r * 34 + k] = Xb[(size_t)(n0 + r) * CFD + k0 + k];
      Bsm[k * 68 + r] = W[(size_t)(o0 + r) * CFD + k0 + k];
    }
    __syncthreads();
#pragma unroll
    for (int kc = 0; kc < 32; kc += 4) {
#pragma unroll
      for (int ti = 0; ti < 2; ti++) {
        v2f a;
        a.x = As[(wm + ti * 16 + l16) * 34 + kc + half * 2];
        a.y = As[(wm + ti * 16 + l16) * 34 + kc + half * 2 + 1];
#pragma unroll
        for (int tj = 0; tj < 2; tj++) {
          v2f bb;
          bb.x = Bsm[(kc + half * 2) * 68 + wn + tj * 16 + l16];
          bb.y = Bsm[(kc + half * 2 + 1) * 68 + wn + tj * 16 + l16];
          acc[ti][tj] = wmma4(a, bb, acc[ti][tj]);
        }
      }
    }
  }
  if (!TRANSOUT) {
#pragma unroll
    for (int ti = 0; ti < 2; ti++)
#pragma unroll
      for (int tj = 0; tj < 2; tj++)
#pragma unroll
        for (int r = 0; r < 8; r++) {
          int gn = n0 + wm + ti * 16 + r + half * 8;
          int go = o0 + wn + tj * 16 + l16;
          float v = acc[ti][tj][r] + bias[go];
          if (RELU) v = fmaxf(v, 0.f);
          OUT[(size_t)(b * NN + gn) * CFD + go] = v;
        }
  } else {
    __syncthreads();
    float* Ct = smem;  // 64 x 65
#pragma unroll
    for (int ti = 0; ti < 2; ti++)
#pragma unroll
      for (int tj = 0; tj < 2; tj++)
#pragma unroll
        for (int r = 0; r < 8; r++) {
          int ln = wm + ti * 16 + r + half * 8;
          int lo = wn + tj * 16 + l16;
          float v = acc[ti][tj][r] + bias[o0 + lo];
          if (RELU) v = fmaxf(v, 0.f);
          Ct[ln * 65 + lo] = v;
        }
    __syncthreads();
    for (int lin = t; lin < 4096; lin += 128) {
      int oo = lin >> 6, nn = lin & 63;
      OUT[(size_t)(b * CFD + o0 + oo) * NN + n0 + nn] = Ct[nn * 65 + oo];
    }
  }
}

// ---------------------------------------------------------------------------
extern "C" void kernel_launch(void* const* d_in, const int* in_sizes, int n_in,
                              void* d_out, int out_size, void* d_ws, size_t ws_size,
                              hipStream_t stream) {
  (void)in_sizes; (void)n_in; (void)out_size; (void)ws_size;
  const float* pred   = (const float*)d_in[0];
  const float* feat   = (const float*)d_in[1];
  const float* w_cls1 = (const float*)d_in[2];
  const float* b_cls1 = (const float*)d_in[3];
  const float* w_cls2 = (const float*)d_in[4];
  const float* b_cls2 = (const float*)d_in[5];
  const float* w_pro1 = (const float*)d_in[6];
  const float* b_pro1 = (const float*)d_in[7];
  const float* w_pro2 = (const float*)d_in[8];
  const float* b_pro2 = (const float*)d_in[9];
  const float* cls_w  = (const float*)d_in[10];

  float* out = (float*)d_out;
  float* predOut = out;                         // [B,19,H,W]
  float* featOut = out + (size_t)BNB * NC * NN; // [B,256,H,W]

  float* ws = (float*)d_ws;
  size_t off = 0;
  float* AFF    = ws + off; off += (size_t)BNB * NN * NN;   // 134 MB, reused
  float* featT  = ws + off; off += (size_t)BNB * NN * CFD;
  float* xnT    = ws + off; off += (size_t)BNB * NN * CFD;
  float* nfT    = ws + off; off += (size_t)BNB * NN * CFD;
  float* hidT   = ws + off; off += (size_t)BNB * NN * CFD;
  float* predT  = ws + off; off += (size_t)BNB * NN * CP;
  float* p3T    = ws + off; off += (size_t)BNB * NN * CP;
  float* spnT   = ws + off; off += (size_t)BNB * NN * CP;
  float* ohT    = ws + off; off += (size_t)BNB * NN * CP;
  float* maskb  = ws + off; off += (size_t)BNB * NN;
  int*   pmaxb  = (int*)(ws + off); off += (size_t)BNB * NN;
  float* rowsum = ws + off; off += (size_t)BNB * NN;
  float* d1     = ws + off; off += (size_t)BNB * NN;
  float* d2     = ws + off; off += (size_t)BNB * NN;

  k_prep<<<dim3((BNB * NN + 255) / 256), dim3(256), 0, stream>>>(pred, cls_w, predT,
                                                                 maskb, pmaxb);
  k_transpose_feat<<<dim3(NN / 32, CFD / 32, BNB), dim3(32, 8), 0, stream>>>(feat, featT);
  k_rownorm<<<dim3(BNB * NN), dim3(64), 0, stream>>>(featT, xnT);
  k_gram_clamp<<<dim3(NN / 64, NN / 64, BNB), dim3(128), 0, stream>>>(xnT, AFF);
  k_topk_threshold<<<dim3(BNB * NN), dim3(256), 0, stream>>>(AFF, rowsum);
  k_colsum_d1<<<dim3((BNB * NN + 255) / 256), dim3(256), 0, stream>>>(AFF, rowsum, d1);
  k_prop_pred<<<dim3(NN / 16, BNB), dim3(32), 0, stream>>>(AFF, predT, d1, maskb, p3T);
  k_cls_head<<<dim3((BNB * NN + 255) / 256), dim3(256), 0, stream>>>(
      p3T, w_cls1, b_cls1, w_cls2, b_cls2, predOut, spnT);
  k_oh<<<dim3((BNB * NN * CP + 255) / 256), dim3(256), 0, stream>>>(pmaxb, ohT);
  k_afff<<<dim3(NN / 64, NN / 64, BNB), dim3(128), 0, stream>>>(spnT, ohT, AFF);
  k_rowsum_rsqrt<<<dim3(BNB * NN), dim3(256), 0, stream>>>(AFF, d2);
  k_prop_feat<<<dim3(NN / 64, CFD / 64, BNB), dim3(128), 0, stream>>>(AFF, featT, d2, nfT);
  k_dense<true, false><<<dim3(NN / 64, CFD / 64, BNB), dim3(128), 0, stream>>>(
      nfT, w_pro1, b_pro1, hidT);
  k_dense<false, true><<<dim3(NN / 64, CFD / 64, BNB), dim3(128), 0, stream>>>(
      hidT, w_pro2, b_pro2, featOut);
}